// FastUpsampleOp_68118181315070
// MI455X (gfx1250) — compile-verified
//
#include <hip/hip_runtime.h>

// FastUpsampleOp: Bayer-pattern 3x nearest upsample + interleave.
// out[b,c,y,x] = in[b,c,row(y),col(x)]  (pure gather, memory-bound)
//   row(p) = p even ? 2*((p/2)/3)+1 : 2*((p/2+2)/3) ; col identical.
// Strategy: per block, async-DMA the needed input rows into LDS
// (GLOBAL_LOAD_ASYNC_TO_LDS_B128, ASYNCcnt), then gather from LDS and emit
// coalesced 16B stores. HBM traffic ~332MB -> ~14.3us floor at 23.3 TB/s.

#define B_  8
#define C_  16
#define H_  256
#define W_  256
#define HO_ (H_*3 - 4)   // 764
#define WO_ (W_*3 - 4)   // 764
#define TILE_Y 8
#define NTHREADS 256
#define QW_ (WO_/4)      // 191 float4 quads per output row

typedef int v4i __attribute__((vector_size(4 * sizeof(int))));
typedef __attribute__((address_space(1))) v4i* gv4i_p;   // global
typedef __attribute__((address_space(3))) v4i* lv4i_p;   // LDS
typedef __attribute__((address_space(3))) void* l_void_p;

__device__ __forceinline__ void async_copy_b128(const float* gsrc, float* lds_dst) {
#if __has_builtin(__builtin_amdgcn_global_load_async_to_lds_b128)
    __builtin_amdgcn_global_load_async_to_lds_b128((gv4i_p)gsrc, (lv4i_p)lds_dst,
                                                   /*offset=*/0, /*cpol=*/0);
#else
    unsigned lofs = (unsigned)(unsigned long long)(l_void_p)lds_dst;
    asm volatile("global_load_async_to_lds_b128 %0, %1, off"
                 :: "v"(lofs), "v"(gsrc) : "memory");
#endif
}

__device__ __forceinline__ void wait_asynccnt0() {
#if __has_builtin(__builtin_amdgcn_s_wait_asynccnt)
    __builtin_amdgcn_s_wait_asynccnt(0);
#else
    asm volatile("s_wait_asynccnt 0" ::: "memory");
#endif
}

// row/col source-index map (identical formula for both axes)
__device__ __forceinline__ int mapIdx(int p) {
    int u = p >> 1;
    int t = (p & 1) ? (u + 2) : u;
    int d = t / 3;                       // magic-multiply, small range
    return (p & 1) ? (2 * d) : (2 * d + 1);
}

__global__ __launch_bounds__(NTHREADS)
void FastUpsampleOp_68118181315070_kernel(const float* __restrict__ in,
                                          float* __restrict__ out) {
    __shared__ __align__(16) float rows[TILE_Y][W_];   // 8 KB staged source rows

    const int    plane = blockIdx.y;                   // b*C + c  (0..127)
    const int    y0    = blockIdx.x * TILE_Y;          // first output row of tile
    const float* inP   = in  + (size_t)plane * (H_ * W_);
    float*       outP  = out + (size_t)plane * ((size_t)HO_ * WO_);
    const int    tid   = threadIdx.x;

    // ---- Stage: async-copy TILE_Y source rows (1KB each) into LDS ----
    // thread t covers 32 bytes of LDS row (t>>5): two b128 async loads.
    {
        const int r = tid >> 5;           // LDS row slot 0..7
        const int l = tid & 31;           // 32-byte chunk within row
        const int y = y0 + r;
        const int srcRow = (y < HO_) ? mapIdx(y) : 0;   // clamp keeps addr valid
        const float* g = inP + srcRow * W_ + l * 8;
        float*       s = &rows[r][l * 8];
        async_copy_b128(g,     s);
        async_copy_b128(g + 4, s + 4);
    }
    wait_asynccnt0();     // our lanes' async copies have landed in LDS
    __syncthreads();      // all waves' copies visible to everyone

    // ---- Compute: gather from LDS, write coalesced float4 quads ----
    for (int idx = tid; idx < TILE_Y * QW_; idx += NTHREADS) {
        const int rr = idx / QW_;           // LDS row slot
        const int q  = idx - rr * QW_;      // quad index within output row
        const int yy = y0 + rr;
        if (yy >= HO_) continue;            // tail tile (y0=760 has 4 rows)
        const int xb = q * 4;
        float4 v;
        v.x = rows[rr][mapIdx(xb + 0)];
        v.y = rows[rr][mapIdx(xb + 1)];
        v.z = rows[rr][mapIdx(xb + 2)];
        v.w = rows[rr][mapIdx(xb + 3)];
        // 764 % 4 == 0 and 764*764 % 4 == 0 -> every quad is 16B aligned
        *reinterpret_cast<float4*>(outP + (size_t)yy * WO_ + xb) = v;
    }
}

extern "C" void kernel_launch(void* const* d_in, const int* in_sizes, int n_in,
                              void* d_out, int out_size, void* d_ws, size_t ws_size,
                              hipStream_t stream) {
    (void)in_sizes; (void)n_in; (void)out_size; (void)d_ws; (void)ws_size;
    const float* x   = (const float*)d_in[0];
    float*       out = (float*)d_out;
    dim3 grid((HO_ + TILE_Y - 1) / TILE_Y,  // 96 row tiles
              B_ * C_);                     // 128 planes
    FastUpsampleOp_68118181315070_kernel<<<grid, NTHREADS, 0, stream>>>(x, out);
}